// SoftPatternClassifier_2207613190521
// MI455X (gfx1250) — compile-verified
//
#include <hip/hip_runtime.h>
#include <hip/hip_bf16.h>
#include <math.h>

typedef __attribute__((ext_vector_type(2))) float v2f;
typedef __attribute__((ext_vector_type(8))) float v8f;

#define NP     64      // patterns P
#define LL     6       // L
#define NTOK   10000   // N
#define DD     300     // D
#define HH     256     // H
#define JD     384     // P*L  (diag rows)
#define JROWS  704     // 384 diag + 320 superdiag rows

// ---------------------------------------------------------------------------
// Kernel 1: gather diag/superdiag rows of w into a packed (704 x 300) slab and
// compute log(norm(row)) per row.
// ---------------------------------------------------------------------------
__global__ void __launch_bounds__(64)
prep_kernel(const float* __restrict__ w, float* __restrict__ Wcat,
            float* __restrict__ lognorm) {
  const int j = blockIdx.x;    // 0..703
  const int t = threadIdx.x;   // 0..63
  const float* src;
  if (j < JD) {
    const int p = j / LL, l = j % LL;
    src = w + (((size_t)p * LL + l) * LL + l) * DD;          // w[p,l,l,:]
  } else {
    const int jj = j - JD;
    const int p = jj / (LL - 1), l = jj % (LL - 1);
    src = w + (((size_t)p * LL + l) * LL + (l + 1)) * DD;    // w[p,l,l+1,:]
  }
  float ss = 0.f;
  for (int d = t; d < DD; d += 64) {
    const float v = src[d];
    Wcat[(size_t)j * DD + d] = v;
    ss += v * v;
  }
  __shared__ float red[64];
  red[t] = ss;
  __syncthreads();
  for (int s = 32; s > 0; s >>= 1) {
    if (t < s) red[t] += red[t + s];
    __syncthreads();
  }
  if (t == 0) lognorm[j] = 0.5f * logf(red[0]);   // log(sqrt(sumsq))
}

// ---------------------------------------------------------------------------
// Kernel 2: Y[n, j] = sigmoid( sum_d emb[doc[n], d] * Wcat[j, d] - lognorm[j] )
// One wave computes a 16(M) x 64(N) tile: A fragment reused across 4 WMMAs.
// Exact f32 tensor path: V_WMMA_F32_16X16X4_F32, K-loop 300/4 = 75 steps.
// ---------------------------------------------------------------------------
__global__ void __launch_bounds__(32)
gemm_wmma_kernel(const int* __restrict__ doc, const float* __restrict__ emb,
                 const float* __restrict__ Wcat,
                 const float* __restrict__ lognorm, float* __restrict__ Y) {
  const int lane = threadIdx.x;   // 0..31
  const int half = lane >> 4;     // selects K sub-pair per ISA 16x4 f32 layout
  const int lid  = lane & 15;
  const int m0 = blockIdx.x * 16;   // token tile   (625 tiles)
  const int j0 = blockIdx.y * 64;   // 4 col tiles  (11 blocks)

  const float* Arow  = emb  + (size_t)doc[m0 + lid] * DD;   // gathered X row
  const float* Brow0 = Wcat + (size_t)(j0      + lid) * DD;
  const float* Brow1 = Wcat + (size_t)(j0 + 16 + lid) * DD;
  const float* Brow2 = Wcat + (size_t)(j0 + 32 + lid) * DD;
  const float* Brow3 = Wcat + (size_t)(j0 + 48 + lid) * DD;

  v8f acc0 = {}, acc1 = {}, acc2 = {}, acc3 = {};

  for (int k = 0; k < DD; k += 4) {
    const int ka = k + 2 * half;               // lanes 0-15: K=k,k+1 ; 16-31: K=k+2,k+3
    const v2f a  = *(const v2f*)(Arow  + ka);  // 8B aligned (doc*1200 and ka even)
    const v2f b0 = *(const v2f*)(Brow0 + ka);
    const v2f b1 = *(const v2f*)(Brow1 + ka);
    const v2f b2 = *(const v2f*)(Brow2 + ka);
    const v2f b3 = *(const v2f*)(Brow3 + ka);
    acc0 = __builtin_amdgcn_wmma_f32_16x16x4_f32(false, a, false, b0, (short)0, acc0, false, false);
    acc1 = __builtin_amdgcn_wmma_f32_16x16x4_f32(false, a, false, b1, (short)0, acc1, false, false);
    acc2 = __builtin_amdgcn_wmma_f32_16x16x4_f32(false, a, false, b2, (short)0, acc2, false, false);
    acc3 = __builtin_amdgcn_wmma_f32_16x16x4_f32(false, a, false, b3, (short)0, acc3, false, false);
  }

  // D layout (16x16 f32): VGPR v -> M = v + 8*half, N = lid.
  const float ln0 = lognorm[j0      + lid];
  const float ln1 = lognorm[j0 + 16 + lid];
  const float ln2 = lognorm[j0 + 32 + lid];
  const float ln3 = lognorm[j0 + 48 + lid];
#pragma unroll
  for (int v = 0; v < 8; ++v) {
    const size_t rbase = (size_t)(m0 + v + 8 * half) * JROWS;
    Y[rbase + j0      + lid] = 1.f / (1.f + expf(ln0 - acc0[v]));
    Y[rbase + j0 + 16 + lid] = 1.f / (1.f + expf(ln1 - acc1[v]));
    Y[rbase + j0 + 32 + lid] = 1.f / (1.f + expf(ln2 - acc2[v]));
    Y[rbase + j0 + 48 + lid] = 1.f / (1.f + expf(ln3 - acc3[v]));
  }
}

// ---------------------------------------------------------------------------
// Kernel 3: per-pattern sequential scan over 10000 tokens.
// h[0] = h[0]*a0 + 1 ; h[l] = h[l]*a_l + h[l-1]*b_{l-1} (old h) ; score += h[5].
// One lane per pattern; cross-iteration critical path = 1 FMA.
// ---------------------------------------------------------------------------
__global__ void __launch_bounds__(64)
scan_kernel(const float* __restrict__ Y, float* __restrict__ scores) {
  const int p = threadIdx.x;                       // 0..63
  const float* Ya = Y + p * LL;                    // a cols p*6 .. p*6+5
  const float* Yb = Y + JD + p * (LL - 1);         // b cols 384+p*5 .. +4
  float h0 = 1.f, h1 = 0.f, h2 = 0.f, h3 = 0.f, h4 = 0.f, h5 = 0.f, sc = 0.f;
  for (int n = 0; n < NTOK; ++n) {
    const float* ya = Ya + (size_t)n * JROWS;
    const float* yb = Yb + (size_t)n * JROWS;
    const v2f a01 = *(const v2f*)(ya);     // 8B aligned: 6p and 704n even
    const v2f a23 = *(const v2f*)(ya + 2);
    const v2f a45 = *(const v2f*)(ya + 4);
    const float b0 = yb[0], b1 = yb[1], b2 = yb[2], b3 = yb[3], b4 = yb[4];
    const float n5 = h5 * a45.y + h4 * b4;   // all n_l use OLD h values
    const float n4 = h4 * a45.x + h3 * b3;
    const float n3 = h3 * a23.y + h2 * b2;
    const float n2 = h2 * a23.x + h1 * b1;
    const float n1 = h1 * a01.y + h0 * b0;
    const float n0 = h0 * a01.x + 1.f;
    h0 = n0; h1 = n1; h2 = n2; h3 = n3; h4 = n4; h5 = n5;
    sc += n5;
  }
  scores[p] = sc;
}

// ---------------------------------------------------------------------------
// Kernel 4: hdn = relu(W1 @ scores + b1) ; out = log_softmax(W2 @ hdn + b2)
// ---------------------------------------------------------------------------
__global__ void __launch_bounds__(256)
mlp_kernel(const float* __restrict__ scores, const float* __restrict__ W1,
           const float* __restrict__ b1, const float* __restrict__ W2,
           const float* __restrict__ b2, float* __restrict__ out) {
  __shared__ float s_sc[NP];
  __shared__ float s_h[HH];
  __shared__ float s_z[2];
  const int t = threadIdx.x;     // 0..255
  if (t < NP) s_sc[t] = scores[t];
  __syncthreads();
  float acc = b1[t];
#pragma unroll 8
  for (int i = 0; i < NP; ++i) acc += W1[t * NP + i] * s_sc[i];   // W1 is (H,P)
  s_h[t] = fmaxf(acc, 0.f);
  __syncthreads();
  if (t < 2) {
    float z = b2[t];
    for (int i = 0; i < HH; ++i) z += W2[t * HH + i] * s_h[i];    // W2 is (C,H)
    s_z[t] = z;
  }
  __syncthreads();
  if (t == 0) {
    const float z0 = s_z[0], z1 = s_z[1];
    const float m  = fmaxf(z0, z1);
    const float lse = m + logf(expf(z0 - m) + expf(z1 - m));
    out[0] = z0 - lse;
    out[1] = z1 - lse;
  }
}

// ---------------------------------------------------------------------------
extern "C" void kernel_launch(void* const* d_in, const int* in_sizes, int n_in,
                              void* d_out, int out_size, void* d_ws, size_t ws_size,
                              hipStream_t stream) {
  const int*   doc = (const int*)  d_in[0];
  const float* emb = (const float*)d_in[1];
  const float* w   = (const float*)d_in[2];
  const float* W1  = (const float*)d_in[3];
  const float* b1  = (const float*)d_in[4];
  const float* W2  = (const float*)d_in[5];
  const float* b2  = (const float*)d_in[6];
  float* out = (float*)d_out;

  char* ws = (char*)d_ws;
  size_t off = 0;
  auto walloc = [&](size_t bytes) -> void* {
    void* p = ws + off;
    off = (off + bytes + 255) & ~(size_t)255;
    return p;
  };
  float* Wcat    = (float*)walloc((size_t)JROWS * DD * sizeof(float)); // 0.85 MB
  float* lognorm = (float*)walloc((size_t)JROWS * sizeof(float));
  float* Y       = (float*)walloc((size_t)NTOK * JROWS * sizeof(float)); // 28.2 MB
  float* scores  = (float*)walloc((size_t)NP * sizeof(float));
  (void)ws_size; (void)in_sizes; (void)n_in; (void)out_size;

  prep_kernel<<<JROWS, 64, 0, stream>>>(w, Wcat, lognorm);
  gemm_wmma_kernel<<<dim3(NTOK / 16, JROWS / 64), 32, 0, stream>>>(doc, emb, Wcat, lognorm, Y);
  scan_kernel<<<1, 64, 0, stream>>>(Y, scores);
  mlp_kernel<<<1, HH, 0, stream>>>(scores, W1, b1, W2, b2, out);
}